// Autoencoder_10823317586732
// MI455X (gfx1250) — compile-verified
//
#include <hip/hip_runtime.h>
#include <stdint.h>

#define Bn 2
#define Hn 512
#define Wn 512
#define Dn 128
#define D3 (Dn*Dn*Dn)          // 2,097,152
#define HW (Hn*Wn)             // 262,144
#define NPIX (Bn*HW)           // 524,288
#define NUM_STEPS 180
#define DTC 0.02f
#define JIT 0.01f

typedef float v2f __attribute__((ext_vector_type(2)));
typedef float v8f __attribute__((ext_vector_type(8)));

// ---- workspace layout (float offsets) ----
#define LEN_OFF  0                     // NPIX floats: raw ray_length
#define MM_OFF   (NPIX)                // 4 u32: [0]=max bits, [1]=min bits
#define RD_OFF   (NPIX + 4)            // 3*NPIX floats: unnormalized ray dirs
#define VOL_OFF  (RD_OFF + 3*NPIX)     // Bn*D3*4 floats: interleaved float4 volume

__device__ inline uint32_t hashu(uint32_t a) {
  a ^= a >> 16; a *= 0x7feb352dU; a ^= a >> 15; a *= 0x846ca68bU; a ^= a >> 16;
  return a;
}
__device__ inline float u01(uint32_t h) { return (float)(h >> 8) * (1.0f / 16777216.0f); }

// ---------------- init min/max cells ----------------
__global__ void init_mm_kernel(unsigned* mm) {
  if (threadIdx.x == 0) { mm[0] = 0u; mm[1] = 0x7F800000u; }
}

// ---------------- planar -> interleaved float4 relayout ----------------
__global__ __launch_bounds__(256) void relayout_kernel(const float* __restrict__ tpl,
                                                       float* __restrict__ vol) {
  int i = blockIdx.x * 256 + threadIdx.x;        // 0 .. Bn*D3-1
  int b = i >> 21;                                // D3 = 2^21
  int v = i & (D3 - 1);
  const float* src = tpl + (size_t)b * 4 * D3 + v;
  float4 o;
  o.x = src[0]; o.y = src[D3]; o.z = src[2 * D3]; o.w = src[3 * D3];
  ((float4*)vol)[i] = o;
}

// ---------------- WMMA ray-direction rotation ----------------
// Each wave computes 16 pixels: D(16x16) = A(16x4: per-pixel [dx,dy,1,0]) x B(4x16: R rows, zero pad)
// rd_k = sum_j R[j][k] * d_j  -> B[j][n] = R[j][n]
__global__ __launch_bounds__(256) void wmma_dirs_kernel(const float* __restrict__ rot,
                                                        const float* __restrict__ foc,
                                                        const float* __restrict__ pp,
                                                        float* __restrict__ rd) {
  int tid  = blockIdx.x * 256 + threadIdx.x;
  int wave = tid >> 5;
  int lane = tid & 31;
  int p0   = wave << 4;                  // 16 pixels per wave
  int lm   = lane & 15;
  int p    = p0 + lm;
  int bi   = p >> 18;                    // HW = 2^18
  int r    = p & (HW - 1);
  int py   = r >> 9;                     // Wn = 512
  int px   = r & (Wn - 1);
  bool hi  = lane >= 16;

  float f0  = foc[bi * 2], f1 = foc[bi * 2 + 1];
  float pp0 = pp[bi * 2],  pp1 = pp[bi * 2 + 1];

  // A: VGPR0 = K0 (lanes<16) / K2 (lanes>=16); VGPR1 = K1 / K3
  v2f a;
  a.x = hi ? 1.0f : ((float)px - pp0) / f0;   // K=0 -> dx ; K=2 -> dz=1
  a.y = hi ? 0.0f : ((float)py - pp1) / f1;   // K=1 -> dy ; K=3 -> 0
  // B: VGPR0 = K0 / K2 rows; VGPR1 = K1 / K3 rows; columns N = lane&15
  const float* R = rot + bi * 9;
  v2f bm;
  bm.x = (lm < 3) ? (hi ? R[2 * 3 + lm] : R[0 * 3 + lm]) : 0.0f;
  bm.y = (!hi && lm < 3) ? R[1 * 3 + lm] : 0.0f;

  v8f c = {};
  c = __builtin_amdgcn_wmma_f32_16x16x4_f32(false, a, false, bm, (short)0, c, false, false);

  // D layout: VGPR r holds M=r (lanes 0-15) / M=8+r (lanes 16-31), N = lane&15
  if (lm < 3) {
    int mbase = hi ? 8 : 0;
#pragma unroll
    for (int rr = 0; rr < 8; ++rr)
      rd[(size_t)(p0 + mbase + rr) * 3 + lm] = c[rr];
  }
}

// ---------------- main ray march ----------------
__global__ __launch_bounds__(256) void march_kernel(
    const float* __restrict__ rot, const float* __restrict__ cpos,
    const float* __restrict__ foc, const float* __restrict__ pp,
    const float* __restrict__ tpl, const float* __restrict__ bg,
    const float* __restrict__ vol, const float* __restrict__ rdbuf,
    float* __restrict__ lenbuf, unsigned* __restrict__ mm,
    float* __restrict__ out, int useVol, int useRd) {
  const int x  = blockIdx.x * 32 + threadIdx.x;
  const int y  = blockIdx.y * 8  + threadIdx.y;
  const int bi = blockIdx.z;
  const int p  = bi * HW + y * Wn + x;

  // ray direction
  float rx, ry, rz;
  if (useRd) {
    rx = rdbuf[3 * p]; ry = rdbuf[3 * p + 1]; rz = rdbuf[3 * p + 2];
  } else {
    float d0 = ((float)x - pp[bi * 2]) / foc[bi * 2];
    float d1 = ((float)y - pp[bi * 2 + 1]) / foc[bi * 2 + 1];
    const float* R = rot + bi * 9;
    rx = R[0] * d0 + R[3] * d1 + R[6];
    ry = R[1] * d0 + R[4] * d1 + R[7];
    rz = R[2] * d0 + R[5] * d1 + R[8];
  }
  float inv = rsqrtf(rx * rx + ry * ry + rz * rz);
  rx *= inv; ry *= inv; rz *= inv;

  const float cx = cpos[bi * 3], cy = cpos[bi * 3 + 1], cz = cpos[bi * 3 + 2];

  // box intersection
  float ix = 1.0f / rx, iy = 1.0f / ry, iz = 1.0f / rz;
  float t1x = (-1.0f - cx) * ix, t2x = (1.0f - cx) * ix;
  float t1y = (-1.0f - cy) * iy, t2y = (1.0f - cy) * iy;
  float t1z = (-1.0f - cz) * iz, t2z = (1.0f - cz) * iz;
  float tmin = fmaxf(fminf(t1x, t2x), fmaxf(fminf(t1y, t2y), fminf(t1z, t2z)));
  float tmax = fminf(fmaxf(t1x, t2x), fminf(fmaxf(t1y, t2y), fmaxf(t1z, t2z)));
  float t = (tmin < tmax) ? fmaxf(tmin, 0.0f) : 0.0f;
  t -= DTC * u01(hashu(0xA511E9B3u ^ (uint32_t)p));

  float px = cx + rx * t, py = cy + ry * t, pz = cz + rz * t;

  float r0 = 0.f, r1 = 0.f, r2 = 0.f, alpha = 0.f, len = 0.f;
  const float4* vb = (const float4*)vol + (size_t)bi * D3;
  const float*  tb = tpl + (size_t)bi * 4 * D3;

  for (int i = 0; i < NUM_STEPS; ++i) {
    // deterministic step jitter: CLT-approx normal from two hashed uniforms
    uint32_t h1 = hashu((uint32_t)p * 1664525u + (uint32_t)i * 1013904223u + 0x9e3779b9u);
    uint32_t h2 = hashu(h1 ^ 0x85ebca6bu);
    float nz = (u01(h1) + u01(h2) - 1.0f) * 2.4494897f;
    float step = DTC * __expf(JIT * nz);

    bool valid = px > -1.f && px < 1.f && py > -1.f && py < 1.f && pz > -1.f && pz < 1.f;
    if (valid && alpha < 1.0f) {
      float g0 = fminf(fmaxf((px + 1.f) * 0.5f * (Dn - 1), 0.f), (float)(Dn - 1));
      float g1 = fminf(fmaxf((py + 1.f) * 0.5f * (Dn - 1), 0.f), (float)(Dn - 1));
      float g2 = fminf(fmaxf((pz + 1.f) * 0.5f * (Dn - 1), 0.f), (float)(Dn - 1));
      int x0 = (int)g0, y0 = (int)g1, z0 = (int)g2;
      int x1 = min(x0 + 1, Dn - 1), y1 = min(y0 + 1, Dn - 1), z1 = min(z0 + 1, Dn - 1);
      float wx = g0 - (float)x0, wy = g1 - (float)y0, wz = g2 - (float)z0;
      float ux = 1.f - wx, uy = 1.f - wy, uz = 1.f - wz;
      int i000 = (z0 * Dn + y0) * Dn + x0, i100 = (z0 * Dn + y0) * Dn + x1;
      int i010 = (z0 * Dn + y1) * Dn + x0, i110 = (z0 * Dn + y1) * Dn + x1;
      int i001 = (z1 * Dn + y0) * Dn + x0, i101 = (z1 * Dn + y0) * Dn + x1;
      int i011 = (z1 * Dn + y1) * Dn + x0, i111 = (z1 * Dn + y1) * Dn + x1;
      float w000 = ux * uy * uz, w100 = wx * uy * uz, w010 = ux * wy * uz, w110 = wx * wy * uz;
      float w001 = ux * uy * wz, w101 = wx * uy * wz, w011 = ux * wy * wz, w111 = wx * wy * wz;

      float s0 = 0.f, s1 = 0.f, s2 = 0.f, s3 = 0.f;
      if (useVol) {
        float4 v;
#define ACC4(W, I) v = vb[I]; s0 += (W)*v.x; s1 += (W)*v.y; s2 += (W)*v.z; s3 += (W)*v.w;
        ACC4(w000, i000) ACC4(w100, i100) ACC4(w010, i010) ACC4(w110, i110)
        ACC4(w001, i001) ACC4(w101, i101) ACC4(w011, i011) ACC4(w111, i111)
#undef ACC4
      } else {
#define ACCP(W, I) s0 += (W)*tb[I]; s1 += (W)*tb[D3+(I)]; s2 += (W)*tb[2*D3+(I)]; s3 += (W)*tb[3*D3+(I)];
        ACCP(w000, i000) ACCP(w100, i100) ACCP(w010, i010) ACCP(w110, i110)
        ACCP(w001, i001) ACCP(w101, i101) ACCP(w011, i011) ACCP(w111, i111)
#undef ACCP
      }
      float contrib = fminf(alpha + s3 * step, 1.0f) - alpha;
      r0 += s0 * contrib; r1 += s1 * contrib; r2 += s2 * contrib;
      alpha += contrib;
      len += (contrib == 0.0f) ? step : 0.0f;
    } else {
      len += step;
    }
    px += rx * step; py += ry * step; pz += rz * step;
  }

  // composite over background + write outputs
  int ob = bi * 3 * HW + y * Wn + x;
  out[ob]          = r0 + (1.f - alpha) * fmaxf(bg[ob], 0.f);
  out[ob + HW]     = r1 + (1.f - alpha) * fmaxf(bg[ob + HW], 0.f);
  out[ob + 2 * HW] = r2 + (1.f - alpha) * fmaxf(bg[ob + 2 * HW], 0.f);
  out[3 * NPIX + p] = alpha;
  lenbuf[p] = len;

  // wave32 reduction of len max/min, then uint-bitcast atomics (len >= 0)
  float vmax = len, vmin = len;
#pragma unroll
  for (int off = 16; off > 0; off >>= 1) {
    vmax = fmaxf(vmax, __shfl_xor(vmax, off));
    vmin = fminf(vmin, __shfl_xor(vmin, off));
  }
  if (threadIdx.x == 0) {
    atomicMax(&mm[0], __float_as_uint(vmax));
    atomicMin(&mm[1], __float_as_uint(vmin));
  }
}

// ---------------- finalize ray_length ----------------
__global__ __launch_bounds__(256) void finalize_kernel(const float* __restrict__ lenbuf,
                                                       const unsigned* __restrict__ mm,
                                                       float* __restrict__ out) {
  int i = blockIdx.x * 256 + threadIdx.x;
  float denom = __uint_as_float(mm[0]) + __uint_as_float(mm[1]);
  float alpha = out[3 * NPIX + i];
  out[4 * NPIX + i] = alpha * lenbuf[i] / denom;
}

extern "C" void kernel_launch(void* const* d_in, const int* in_sizes, int n_in,
                              void* d_out, int out_size, void* d_ws, size_t ws_size,
                              hipStream_t stream) {
  (void)in_sizes; (void)n_in; (void)out_size;
  const float* rot  = (const float*)d_in[0];
  const float* cpos = (const float*)d_in[1];
  const float* foc  = (const float*)d_in[2];
  const float* pp   = (const float*)d_in[3];
  // d_in[4] = pixel_coords (pure meshgrid, recomputed analytically)
  const float* tpl  = (const float*)d_in[5];
  const float* bg   = (const float*)d_in[6];
  float* out = (float*)d_out;
  float* ws  = (float*)d_ws;

  size_t need_rd  = (size_t)(RD_OFF + 3 * NPIX) * sizeof(float);
  size_t need_vol = (size_t)(VOL_OFF + (size_t)Bn * D3 * 4) * sizeof(float);
  int useRd  = ws_size >= need_rd;
  int useVol = ws_size >= need_vol;

  init_mm_kernel<<<1, 64, 0, stream>>>((unsigned*)(ws + MM_OFF));
  if (useVol)
    relayout_kernel<<<(Bn * D3) / 256, 256, 0, stream>>>(tpl, ws + VOL_OFF);
  if (useRd)
    wmma_dirs_kernel<<<(NPIX / 16 * 32) / 256, 256, 0, stream>>>(rot, foc, pp, ws + RD_OFF);

  dim3 blk(32, 8, 1), grd(Wn / 32, Hn / 8, Bn);
  march_kernel<<<grd, blk, 0, stream>>>(rot, cpos, foc, pp, tpl, bg,
                                        ws + VOL_OFF, ws + RD_OFF, ws + LEN_OFF,
                                        (unsigned*)(ws + MM_OFF), out, useVol, useRd);

  finalize_kernel<<<NPIX / 256, 256, 0, stream>>>(ws + LEN_OFF,
                                                  (const unsigned*)(ws + MM_OFF), out);
}